// GENConv_3040836846116
// MI455X (gfx1250) — compile-verified
//
#include <hip/hip_runtime.h>
#include <math.h>

typedef __attribute__((ext_vector_type(2))) float v2f;
typedef __attribute__((ext_vector_type(8))) float v8f;

#define N_NODES 100000
#define N_EDGES 1200000
#define DIM     64
#define EPS     1e-7f

// ---------------------------------------------------------------------------
// Kernel 1: per-destination-node edge aggregation (dst is sorted -> each
// node's incoming edges are contiguous). One wave32 per node, 2 channels/lane.
// Online (flash) per-channel softmax: single pass over edges, deterministic,
// no atomics, no [E,D] materialization. Then MessageNorm + residual, storing
// normalized feats into workspace for the WMMA GEMM.
// ---------------------------------------------------------------------------
__global__ __launch_bounds__(256) void genconv_edge_kernel(
    const float* __restrict__ nodef,   // [N, 64]
    const float* __restrict__ edgef,   // [E, 64]
    const float* __restrict__ scale,   // [1]
    const int*   __restrict__ src,     // [E]
    const int*   __restrict__ dst,     // [E] sorted
    float*       __restrict__ feats)   // [N, 64] output of phase 1
{
  const int lane = threadIdx.x & 31;
  const int wave = threadIdx.x >> 5;
  const int node = blockIdx.x * 8 + wave;
  if (node >= N_NODES) return;

  // Binary search for edge range [start, end) with dst == node.
  // Uniform across the wave -> effectively scalar.
  int lo = 0, hi = N_EDGES;
  while (lo < hi) { int mid = (lo + hi) >> 1; if (dst[mid] <  node) lo = mid + 1; else hi = mid; }
  const int start = lo;
  hi = N_EDGES;
  while (lo < hi) { int mid = (lo + hi) >> 1; if (dst[mid] <= node) lo = mid + 1; else hi = mid; }
  const int end = lo;

  const int c0 = lane * 2;           // this lane's two channels
  const float NEG_INF = -__builtin_inff();
  float M0 = NEG_INF, M1 = NEG_INF;  // running channel max
  float S0 = 0.f, S1 = 0.f;          // running sum of exp(z - M)
  float T0 = 0.f, T1 = 0.f;          // running sum of m * exp(z - M)

  for (int e = start; e < end; ++e) {
    const int s = src[e];
    const v2f hs = *(const v2f*)(nodef + (size_t)s * DIM + c0);   // random gather
    const v2f ef = *(const v2f*)(edgef + (size_t)e * DIM + c0);   // streaming
    // m = relu(h[src] + ef) + eps ; z = m * BETA with BETA == 1
    const float m0 = fmaxf(hs.x + ef.x, 0.f) + EPS;
    const float m1 = fmaxf(hs.y + ef.y, 0.f) + EPS;
    // online softmax update, channel 0
    const float n0 = fmaxf(M0, m0);
    const float r0 = __expf(M0 - n0);   // exp(-inf) == 0 handles first edge
    const float w0 = __expf(m0 - n0);
    S0 = S0 * r0 + w0;  T0 = T0 * r0 + m0 * w0;  M0 = n0;
    // channel 1
    const float n1 = fmaxf(M1, m1);
    const float r1 = __expf(M1 - n1);
    const float w1 = __expf(m1 - n1);
    S1 = S1 * r1 + w1;  T1 = T1 * r1 + m1 * w1;  M1 = n1;
  }

  // agg = sum(m * softmax_weight) = T / S ; zero-degree -> 0 (matches ref guard)
  const float a0 = (S0 > 0.f) ? T0 / S0 : 0.f;
  const float a1 = (S1 > 0.f) ? T1 / S1 : 0.f;

  // MessageNorm: msg = agg/||agg|| * ||h_n|| * scale ; feats = h_n + msg
  const v2f hn = *(const v2f*)(nodef + (size_t)node * DIM + c0);
  float asq = a0 * a0 + a1 * a1;
  float hsq = hn.x * hn.x + hn.y * hn.y;
  #pragma unroll
  for (int off = 16; off > 0; off >>= 1) {
    asq += __shfl_xor(asq, off, 32);
    hsq += __shfl_xor(hsq, off, 32);
  }
  const float k = (scale[0] * sqrtf(hsq)) / fmaxf(sqrtf(asq), 1e-12f);

  v2f o;
  o.x = hn.x + a0 * k;
  o.y = hn.y + a1 * k;
  *(v2f*)(feats + (size_t)node * DIM + c0) = o;
}

// ---------------------------------------------------------------------------
// Kernel 2: out = feats @ W^T + b  via V_WMMA_F32_16X16X4_F32.
// One wave per 16-row tile; 4 column tiles (full 64 cols) accumulated in
// 4 x v8f. K swept 64/4 = 16 steps. B[k][j] = W[j][k], so both A and B
// fragments are contiguous float2 loads per lane (ISA 16x4 f32 layout:
// lanes 0-15 carry K = k,k+1 ; lanes 16-31 carry K = k+2,k+3).
// ---------------------------------------------------------------------------
__global__ __launch_bounds__(256) void genconv_gemm_kernel(
    const float* __restrict__ feats,   // [N, 64]
    const float* __restrict__ W,       // [64, 64] row-major, W[j][k]
    const float* __restrict__ bias,    // [64]
    float*       __restrict__ out)     // [N, 64]
{
  const int lane = threadIdx.x & 31;
  const int wave = threadIdx.x >> 5;
  const int tile = blockIdx.x * 8 + wave;              // wave-uniform
  const int NT = N_NODES / 16;                         // 6250, exact
  if (tile >= NT) return;                              // whole-wave exit: EXEC stays all-1s

  const int row0   = tile * 16;
  const int mrow   = lane & 15;          // M (for A) / N (for B) index
  const int khalf  = (lane >> 4) * 2;    // 0 for lanes 0-15, 2 for lanes 16-31
  const int rowoff = (lane >> 4) * 8;    // C/D layout: upper lanes hold M+8

  v8f acc0 = {}, acc1 = {}, acc2 = {}, acc3 = {};

  #pragma unroll
  for (int kt = 0; kt < 16; ++kt) {
    const int k0 = kt * 4 + khalf;
    const v2f a  = *(const v2f*)(feats + (size_t)(row0 + mrow) * DIM + k0);
    const v2f b0 = *(const v2f*)(W + (size_t)( 0 + mrow) * DIM + k0);
    const v2f b1 = *(const v2f*)(W + (size_t)(16 + mrow) * DIM + k0);
    const v2f b2 = *(const v2f*)(W + (size_t)(32 + mrow) * DIM + k0);
    const v2f b3 = *(const v2f*)(W + (size_t)(48 + mrow) * DIM + k0);
    acc0 = __builtin_amdgcn_wmma_f32_16x16x4_f32(false, a, false, b0, (short)0, acc0, false, false);
    acc1 = __builtin_amdgcn_wmma_f32_16x16x4_f32(false, a, false, b1, (short)0, acc1, false, false);
    acc2 = __builtin_amdgcn_wmma_f32_16x16x4_f32(false, a, false, b2, (short)0, acc2, false, false);
    acc3 = __builtin_amdgcn_wmma_f32_16x16x4_f32(false, a, false, b3, (short)0, acc3, false, false);
  }

  // bias + store (C/D: VGPR r holds row row0 + rowoff + r, col = jt*16 + mrow)
  const float bj0 = bias[ 0 + mrow];
  const float bj1 = bias[16 + mrow];
  const float bj2 = bias[32 + mrow];
  const float bj3 = bias[48 + mrow];
  #pragma unroll
  for (int r = 0; r < 8; ++r) {
    const size_t rowbase = (size_t)(row0 + rowoff + r) * DIM;
    out[rowbase +  0 + mrow] = acc0[r] + bj0;
    out[rowbase + 16 + mrow] = acc1[r] + bj1;
    out[rowbase + 32 + mrow] = acc2[r] + bj2;
    out[rowbase + 48 + mrow] = acc3[r] + bj3;
  }
}

extern "C" void kernel_launch(void* const* d_in, const int* in_sizes, int n_in,
                              void* d_out, int out_size, void* d_ws, size_t ws_size,
                              hipStream_t stream) {
  const float* nodef = (const float*)d_in[0];   // [N, 64]
  const float* edgef = (const float*)d_in[1];   // [E, 64]
  const float* W     = (const float*)d_in[2];   // [64, 64]
  const float* bias  = (const float*)d_in[3];   // [64]
  const float* scale = (const float*)d_in[4];   // [1]
  const int*   src   = (const int*)d_in[5];     // [E]
  const int*   dst   = (const int*)d_in[6];     // [E] sorted

  float* feats = (float*)d_ws;                  // [N, 64] scratch (25.6 MB)
  float* out   = (float*)d_out;                 // [N, 64]

  // Phase 1: 8 waves/block, one node per wave.
  const int blocks1 = (N_NODES + 7) / 8;        // 12500
  genconv_edge_kernel<<<blocks1, 256, 0, stream>>>(nodef, edgef, scale, src, dst, feats);

  // Phase 2: 8 row-tiles per block.
  const int ntiles  = N_NODES / 16;             // 6250
  const int blocks2 = (ntiles + 7) / 8;         // 782
  genconv_gemm_kernel<<<blocks2, 256, 0, stream>>>(feats, W, bias, out);
}